// C3_67946382623525
// MI455X (gfx1250) — compile-verified
//
#include <hip/hip_runtime.h>

// LeNet C3 as implicit GEMM on CDNA5 WMMA (exact fp32, 16x16x4).
// One workgroup (8 waves) per (batch, output-row-PAIR): waves 0-3 -> row oh0,
// waves 4-7 -> row oh0+1; each wave computes D[16oc x 128px] (8 acc tiles).
// Input rows staged once per pair via GLOBAL_LOAD_ASYNC_TO_LDS_B128.

typedef __attribute__((ext_vector_type(2))) float v2f;
typedef __attribute__((ext_vector_type(8))) float v8f;
typedef int i4v __attribute__((vector_size(16)));        // int4 vector for async builtin
typedef __attribute__((address_space(1))) i4v* g_i4p;    // global int4*
typedef __attribute__((address_space(3))) i4v* l_i4p;    // LDS int4*

#define BATCH    32
#define CIN      6
#define HW       512
#define XSW      520      // staged row width: 512 cols + 8 zero-pad (no clamping)
#define OC       16
#define OHW      508
#define KTOT     150      // 6*5*5
#define KPAD     152      // pad K to multiple of 4 (zero A columns)
#define NTHREADS 256      // 8 waves (wave32)
#define NROWS    36       // 6 channels x 6 input rows (covers output rows oh0, oh0+1)

#if defined(__has_builtin)
#if __has_builtin(__builtin_amdgcn_global_load_async_to_lds_b128) && \
    __has_builtin(__builtin_amdgcn_s_wait_asynccnt)
#define HAVE_ASYNC_LDS 1
#endif
#endif
#ifndef HAVE_ASYNC_LDS
#define HAVE_ASYNC_LDS 0
#endif

// LeNet-5 C3 connectivity: bit ic set if input map ic feeds output map oc.
__device__ __constant__ unsigned char conn_mask[16] = {
    0x07, 0x0E, 0x1C, 0x38, 0x31, 0x23,
    0x0F, 0x1E, 0x3C, 0x39, 0x33, 0x27, 0x1B, 0x36, 0x2D,
    0x3F
};

__global__ __launch_bounds__(NTHREADS) void c3_wmma_kernel(
    const float* __restrict__ x,      // [32,6,512,512]
    const float* __restrict__ w,      // [16,6,5,5]
    const float* __restrict__ bias,   // [16]
    float* __restrict__ out)          // [32,16,508,508]
{
    __shared__ float As[KPAD * 16];                   // As[k*16 + m], zero-padded k>=150
    __shared__ __align__(16) int tabb[KPAD];          // byte offset of (ic,kh,kw) in xs (row 0)
    __shared__ __align__(16) float xs[NROWS * XSW];   // xs[(ic*6 + kh)*XSW + col]

    const int tid = threadIdx.x;
    const int n   = blockIdx.x / (OHW / 2);
    const int oh0 = (blockIdx.x % (OHW / 2)) * 2;     // row pair: oh0, oh0+1

    // ---- stage masked weights, k-major (A^T) ----
    for (int j = tid; j < KPAD * 16; j += NTHREADS) {
        int k = j >> 4, m = j & 15;
        float v = 0.0f;
        if (k < KTOT) {
            int ic = k / 25;
            float msk = ((conn_mask[m] >> ic) & 1) ? 1.0f : 0.0f;
            v = w[m * KTOT + k] * msk;   // w[m][ic][kh][kw] flat == w[m*150 + k]
        }
        As[k * 16 + m] = v;
    }
    // ---- per-k byte-offset table (row-parity added via xs base per wave) ----
    for (int k = tid; k < KPAD; k += NTHREADS) {
        int off = 0;
        if (k < KTOT) {
            int ic = k / 25;
            int r  = k - ic * 25;
            int kh = r / 5;
            int kw = r - kh * 5;
            off = ((ic * 6 + kh) * XSW + kw) * 4;
        }
        tabb[k] = off;
    }
    // ---- zero the 8-column pad tail of each staged row ----
    for (int j = tid; j < NROWS * 8; j += NTHREADS) {
        int row = j >> 3, c = j & 7;
        xs[row * XSW + HW + c] = 0.0f;
    }
    // ---- stage 36 input rows (6 ch x 6 kh) for rows oh0..oh0+5 ----
    for (int j = tid; j < NROWS * (HW / 4); j += NTHREADS) {
        int row = j >> 7;                 // 0..35 == ic*6 + kh
        int c4  = j & (HW / 4 - 1);
        int ic  = row / 6;
        int kh  = row - ic * 6;
        const float4* src =
            (const float4*)(x + (((size_t)n * CIN + ic) * HW + (size_t)(oh0 + kh)) * HW) + c4;
        float4* dst = (float4*)(xs + row * XSW) + c4;   // row*XSW*4 bytes is 16B aligned
#if HAVE_ASYNC_LDS
        __builtin_amdgcn_global_load_async_to_lds_b128(
            (g_i4p)(uintptr_t)src, (l_i4p)dst, /*offset=*/0, /*cpol=*/0);
#else
        *dst = *src;
#endif
    }
#if HAVE_ASYNC_LDS
    __builtin_amdgcn_s_wait_asynccnt(0);
#endif
    __syncthreads();

    // ---- per-wave GEMM: D[16 x 128] for output row oh0+p ----
    const int  lane = tid & 31;
    const int  wave = tid >> 5;
    const int  p    = wave >> 2;          // 0..1: output row parity
    const int  q    = wave & 3;           // 0..3: 128-px quarter
    const int  m    = lane & 15;          // A: M index / B,C,D: N index
    const bool hiH  = (lane >= 16);       // upper half-wave: K+2/K+3 (A,B), M+8 (C/D)
    const int  ow0  = q * 128;            // 4 waves x 128 px = 512 >= 508
    const int  cbpx = ow0 + m;
    const char* xsb = (const char*)(xs + p * XSW);   // fold row parity into base

    v8f acc[8];
#pragma unroll
    for (int t = 0; t < 8; ++t) acc[t] = v8f{};

    for (int k0 = 0; k0 < KPAD; k0 += 4) {
        const int klo = k0 + (hiH ? 2 : 0);       // always even -> int2 load aligned
        v2f a;                                    // A frag: VGPR0=A[m,klo], VGPR1=A[m,klo+1]
        a.x = As[klo * 16 + m];
        a.y = As[klo * 16 + 16 + m];
        const int2 off = *(const int2*)(tabb + klo);
        const float* p0 = (const float*)(xsb + off.x) + cbpx;
        const float* p1 = (const float*)(xsb + off.y) + cbpx;
#pragma unroll
        for (int t = 0; t < 8; ++t) {
            v2f b;
            b.x = p0[16 * t];                     // ds_load offset:64*t
            b.y = p1[16 * t];
            acc[t] = __builtin_amdgcn_wmma_f32_16x16x4_f32(
                false, a, false, b, (short)0, acc[t], false, false);
        }
    }

    // ---- epilogue: + bias, masked nontemporal store (output is write-once) ----
    float bv[8];
#pragma unroll
    for (int r = 0; r < 8; ++r) bv[r] = bias[r + (hiH ? 8 : 0)];

    const int oh = oh0 + p;
#pragma unroll
    for (int t = 0; t < 8; ++t) {
        int ow = ow0 + 16 * t + m;
        if (ow < OHW) {
#pragma unroll
            for (int r = 0; r < 8; ++r) {
                int oc = r + (hiH ? 8 : 0);
                __builtin_nontemporal_store(
                    acc[t][r] + bv[r],
                    out + (((size_t)n * OC + oc) * OHW + oh) * OHW + ow);
            }
        }
    }
}

extern "C" void kernel_launch(void* const* d_in, const int* in_sizes, int n_in,
                              void* d_out, int out_size, void* d_ws, size_t ws_size,
                              hipStream_t stream) {
    const float* x    = (const float*)d_in[0];
    const float* w    = (const float*)d_in[1];
    const float* bias = (const float*)d_in[2];
    float* out        = (float*)d_out;

    dim3 grid(BATCH * (OHW / 2));   // 8128 workgroups: one per (batch, output row pair)
    dim3 block(NTHREADS);
    c3_wmma_kernel<<<grid, block, 0, stream>>>(x, w, bias, out);
}